// VectorQuantizer_61005715472983
// MI455X (gfx1250) — compile-verified
//
#include <hip/hip_runtime.h>
#include <stdint.h>

typedef float v2f __attribute__((ext_vector_type(2)));
typedef float v8f __attribute__((ext_vector_type(8)));

#define KCODES 512
#define CDIM   64
#define HWSZ   4096            // 64*64
#define NTASK  4096            // 131072 pixels / 32 per wave
#define NELEM  8388608.0f      // 32*64*64*64
#define ROWP   68              // padded LDS row stride (floats) -> no bank conflicts

// ---------------------------------------------------------------------------
// Kernel 0: zero the used-code mask (ws is not re-poisoned between replays)
// ---------------------------------------------------------------------------
__global__ void vq_zero_kernel(int* __restrict__ used) {
    used[threadIdx.x] = 0;     // launched with 512 threads
}

// ---------------------------------------------------------------------------
// Kernel 1: fused GEMM (f32 WMMA) + argmin + gather + loss partials.
//   grid = 512 blocks x 256 threads (8 waves). Wave w of block b owns 32
//   consecutive pixels. Codebook slabs (16 codes x 64ch = 4KB) are staged
//   into LDS with global_load_async_to_lds_b128, double buffered.
// ---------------------------------------------------------------------------
__global__ __launch_bounds__(256) void vq_main_kernel(
        const float* __restrict__ x,      // [32,64,64,64] = [b][c][h][w]
        const float* __restrict__ cb,     // [512,64]
        float*       __restrict__ out,    // [32,64,64,64]
        int*         __restrict__ used,   // [512]
        float*       __restrict__ partial)// [4096] per-task loss sums
{
    __shared__ float w2[KCODES];
    __shared__ __align__(16) float slab[2][16 * ROWP];
    __shared__ int   idxLds[8 * 32];

    const int tid = threadIdx.x;

    // --- precompute ||w_k||^2 once per workgroup (uniform control flow) ---
    for (int k = tid; k < KCODES; k += 256) {
        const float* row = cb + k * CDIM;
        float s = 0.f;
        #pragma unroll
        for (int c = 0; c < CDIM; ++c) s = fmaf(row[c], row[c], s);
        w2[k] = s;
    }

    // --- async staging: thread tid moves 16B of the 4KB slab for tile t ---
    const int srow = tid >> 4;   // 0..15 : code row within slab
    const int sseg = tid & 15;   // 0..15 : 16B segment within row
    #define STAGE(T, BUF)                                                     \
    {                                                                         \
        uint64_t g = (uint64_t)(uintptr_t)(cb + (size_t)((T) * 16 + srow) * CDIM + sseg * 4); \
        uint32_t l = (uint32_t)(uintptr_t)&slab[(BUF)][srow * ROWP + sseg * 4];\
        asm volatile("global_load_async_to_lds_b128 %0, %1, off"              \
                     :: "v"(l), "v"(g) : "memory");                           \
    }

    STAGE(0, 0);                         // prefetch first slab (ASYNCcnt)

    const int lane = tid & 31;
    const int wave = tid >> 5;
    const int hi   = lane >> 4;          // half-wave: 0 or 1
    const int lp   = lane & 15;

    const int task      = blockIdx.x * 8 + wave;   // 0..4095
    const int pixelBase = task * 32;
    const int bimg      = pixelBase >> 12;         // / 4096
    const int hwB       = pixelBase & 4095;        // 32-aligned, same image row
    const float* xBase  = x + (size_t)bimg * CDIM * HWSZ + hwB;

    // --- A tiles: pixels p0 = lp, p1 = 16+lp ; channels 4kk + 2hi + {0,1} ---
    v2f a0[16], a1[16];
    #pragma unroll
    for (int kk = 0; kk < 16; ++kk) {
        const int c0 = 4 * kk + 2 * hi;
        a0[kk][0] = xBase[(size_t)c0       * HWSZ + lp];
        a0[kk][1] = xBase[(size_t)(c0 + 1) * HWSZ + lp];
        a1[kk][0] = xBase[(size_t)c0       * HWSZ + 16 + lp];
        a1[kk][1] = xBase[(size_t)(c0 + 1) * HWSZ + 16 + lp];
    }

    float best0[8], best1[8];
    int   bi0[8],   bi1[8];
    #pragma unroll
    for (int s = 0; s < 8; ++s) {
        best0[s] = 3.4e38f; best1[s] = 3.4e38f; bi0[s] = 0; bi1[s] = 0;
    }

    asm volatile("s_wait_asynccnt 0x0" ::: "memory");
    __syncthreads();                     // slab(0) + w2 visible to all waves

    // --- sweep 32 column tiles; 32 wmma per tile, B from LDS, dbl-buffered ---
    for (int t = 0; t < 32; ++t) {
        if (t + 1 < 32) STAGE(t + 1, (t + 1) & 1);   // overlap copy w/ compute

        const float* bl = &slab[t & 1][lp * ROWP + 2 * hi];
        v8f acc0 = {}, acc1 = {};
        #pragma unroll
        for (int kk = 0; kk < 16; ++kk) {
            const v2f bv = *(const v2f*)(bl + 4 * kk);     // ds_load_b64
            acc0 = __builtin_amdgcn_wmma_f32_16x16x4_f32(
                       false, a0[kk], false, bv, (short)0, acc0, false, false);
            acc1 = __builtin_amdgcn_wmma_f32_16x16x4_f32(
                       false, a1[kk], false, bv, (short)0, acc1, false, false);
        }

        const int   n   = t * 16 + lp;
        const float w2n = w2[n];
        #pragma unroll
        for (int s = 0; s < 8; ++s) {
            const float d0 = fmaf(-2.f, acc0[s], w2n);  // x^2 const per pixel
            const float d1 = fmaf(-2.f, acc1[s], w2n);
            if (d0 < best0[s]) { best0[s] = d0; bi0[s] = n; }
            if (d1 < best1[s]) { best1[s] = d1; bi1[s] = n; }
        }

        asm volatile("s_wait_asynccnt 0x0" ::: "memory");
        __syncthreads();                 // slab(t+1) staged by all waves
    }

    // --- argmin across the 16 lanes of each half ---
    #pragma unroll
    for (int s = 0; s < 8; ++s) {
        #pragma unroll
        for (int m = 1; m <= 8; m <<= 1) {
            float od; int oi;
            od = __shfl_xor(best0[s], m, 32); oi = __shfl_xor(bi0[s], m, 32);
            if (od < best0[s] || (od == best0[s] && oi < bi0[s])) { best0[s] = od; bi0[s] = oi; }
            od = __shfl_xor(best1[s], m, 32); oi = __shfl_xor(bi1[s], m, 32);
            if (od < best1[s] || (od == best1[s] && oi < bi1[s])) { best1[s] = od; bi1[s] = oi; }
        }
    }

    // --- publish the 32 tile indices through LDS ---
    if (lp == 0) {       // lanes 0 (pixels s+0 / 16+s) and 16 (pixels 8+s / 24+s)
        #pragma unroll
        for (int s = 0; s < 8; ++s) {
            idxLds[wave * 32 + hi * 8 + s]      = bi0[s];
            idxLds[wave * 32 + 16 + hi * 8 + s] = bi1[s];
        }
    }
    __syncthreads();
    const int myIdx0 = idxLds[wave * 32 + lp];
    const int myIdx1 = idxLds[wave * 32 + 16 + lp];

    if (hi == 0) { used[myIdx0] = 1; used[myIdx1] = 1; }  // lanes 0..15 cover all

    // --- gather quantized vectors, write out, accumulate (x - q)^2 ---
    float*       outBase = out + (size_t)bimg * CDIM * HWSZ + hwB;
    const float* q0r     = cb + (size_t)myIdx0 * CDIM + 2 * hi;
    const float* q1r     = cb + (size_t)myIdx1 * CDIM + 2 * hi;
    float lsum = 0.f;
    #pragma unroll
    for (int kk = 0; kk < 16; ++kk) {
        const v2f q0 = *(const v2f*)(q0r + 4 * kk);
        const v2f q1 = *(const v2f*)(q1r + 4 * kk);
        const int c0 = 4 * kk + 2 * hi;
        outBase[(size_t)c0       * HWSZ + lp]      = q0[0];
        outBase[(size_t)(c0 + 1) * HWSZ + lp]      = q0[1];
        outBase[(size_t)c0       * HWSZ + 16 + lp] = q1[0];
        outBase[(size_t)(c0 + 1) * HWSZ + 16 + lp] = q1[1];
        float d;
        d = a0[kk][0] - q0[0]; lsum = fmaf(d, d, lsum);
        d = a0[kk][1] - q0[1]; lsum = fmaf(d, d, lsum);
        d = a1[kk][0] - q1[0]; lsum = fmaf(d, d, lsum);
        d = a1[kk][1] - q1[1]; lsum = fmaf(d, d, lsum);
    }
    #pragma unroll
    for (int m = 1; m < 32; m <<= 1) lsum += __shfl_xor(lsum, m, 32);
    if (lane == 0) partial[task] = lsum;
}

// ---------------------------------------------------------------------------
// Kernel 2: deterministic fixed-order reduction -> losses + unique count
// ---------------------------------------------------------------------------
__global__ __launch_bounds__(256) void vq_final_kernel(
        const float* __restrict__ partial,  // [4096]
        const int*   __restrict__ used,     // [512]
        float*       __restrict__ scalars)  // out + 8388608: [commit, cbk, uniq]
{
    __shared__ float fs[256];
    __shared__ int   is[256];
    const int t = threadIdx.x;
    float s = 0.f;
    for (int i = 0; i < 16; ++i) s += partial[t * 16 + i];
    fs[t] = s;
    is[t] = used[t] + used[t + 256];
    __syncthreads();
    for (int w = 128; w > 0; w >>= 1) {
        if (t < w) { fs[t] += fs[t + w]; is[t] += is[t + w]; }
        __syncthreads();
    }
    if (t == 0) {
        const float mean = fs[0] / NELEM;
        scalars[0] = mean;          // commitment_loss
        scalars[1] = mean;          // codebook_loss (numerically identical)
        scalars[2] = (float)is[0];  // unique_codes
    }
}

// ---------------------------------------------------------------------------
extern "C" void kernel_launch(void* const* d_in, const int* in_sizes, int n_in,
                              void* d_out, int out_size, void* d_ws, size_t ws_size,
                              hipStream_t stream) {
    const float* x  = (const float*)d_in[0];   // [32,64,64,64] f32
    const float* cb = (const float*)d_in[1];   // [512,64] f32
    float* out = (float*)d_out;                // 8388608 + 3 floats

    int*   used    = (int*)d_ws;               // 512 ints
    float* partial = (float*)d_ws + KCODES;    // 4096 floats

    vq_zero_kernel <<<1,   512, 0, stream>>>(used);
    vq_main_kernel <<<512, 256, 0, stream>>>(x, cb, out, used, partial);
    vq_final_kernel<<<1,   256, 0, stream>>>(partial, used, out + 8388608);
}